// NormalizedWeightedFMLayer_35734127902748
// MI455X (gfx1250) — compile-verified
//
#include <hip/hip_runtime.h>
#include <math.h>

typedef __attribute__((ext_vector_type(2))) float v2f;
typedef __attribute__((ext_vector_type(8))) float v8f;

#define FS 26          // sparse fields
#define FD 13          // dense fields
#define NF 39          // total fields
#define EMB_D 4        // embedding dim
#define NSQ (NF*NF)    // 1521
#define BN_EPS 1e-3f

// workspace layout (floats):
//   [0       .. NSQ-1 ]  stats_sum  (Σ_b prod)
//   [NSQ     .. 2NSQ-1]  stats_sq   (Σ_b prod^2)
//   [2NSQ    .. 3NSQ-1]  G          (tanh(beta)/sqrt(var+eps), zero elsewhere)
//   [3NSQ            ]   Cconst     (Σ_p g_p * mean_p)
#define WS_FLOATS (3*NSQ + 1)

// ---------------------------------------------------------------------------
__global__ void fm_zero_kernel(float* ws) {
    int i = blockIdx.x * blockDim.x + threadIdx.x;
    if (i < WS_FLOATS) ws[i] = 0.0f;
}

// ---------------------------------------------------------------------------
// Load the shared A/B operand for 16-field group t of sample b.
// Lane layout for V_WMMA_F32_16X16X4_F32 (A 16x4 and B 4x16 coincide when
// B = A^T): lane L holds row/col (L&15), K = {0,1} (lanes 0-15) or {2,3}.
__device__ inline v2f load_operand(const int* __restrict__ sparse_ids,
                                   const float* __restrict__ dense_vals,
                                   const float* __restrict__ emb_table,
                                   float wpre, int b, int f, int k0, int V) {
    v2f v;
    if (f < FS) {
        int id = sparse_ids[(size_t)b * FS + f];
        const float* e = emb_table + (((size_t)f * V + id) * EMB_D + k0);
        v.x = e[0]; v.y = e[1];
    } else if (f < NF) {
        float x = dense_vals[(size_t)b * FD + (f - FS)] * wpre;
        v.x = x; v.y = x;          // dense embedding constant across D
    } else {
        v.x = 0.0f; v.y = 0.0f;    // padding fields 39..47
    }
    return v;
}

// ---------------------------------------------------------------------------
// Pass 1: per-pair Σ prod and Σ prod^2 over the batch, via fp32 WMMA Gram tiles.
__global__ void fm_stats_kernel(const int* __restrict__ sparse_ids,
                                const float* __restrict__ dense_vals,
                                const float* __restrict__ emb_table,
                                const float* __restrict__ weight,
                                float* __restrict__ stats_sum,
                                float* __restrict__ stats_sq,
                                int B, int V) {
    const int lane   = threadIdx.x & 31;
    const int half   = lane >> 4;
    const int n      = lane & 15;
    const int k0     = half * 2;
    const int wave   = (blockIdx.x * blockDim.x + threadIdx.x) >> 5;
    const int nwaves = (gridDim.x * blockDim.x) >> 5;

    float wpre[3];
#pragma unroll
    for (int t = 0; t < 3; ++t) {
        int f = t * 16 + n;
        wpre[t] = (f >= FS && f < NF) ? weight[f - FS] : 0.0f;
    }

    float accS[6][8], accQ[6][8];
#pragma unroll
    for (int t = 0; t < 6; ++t)
#pragma unroll
        for (int q = 0; q < 8; ++q) { accS[t][q] = 0.0f; accQ[t][q] = 0.0f; }

    for (int b = wave; b < B; b += nwaves) {
        v2f op[3];
#pragma unroll
        for (int t = 0; t < 3; ++t)
            op[t] = load_operand(sparse_ids, dense_vals, emb_table,
                                 wpre[t], b, t * 16 + n, k0, V);
        int ti = 0;
#pragma unroll
        for (int r = 0; r < 3; ++r)
#pragma unroll
            for (int c = r; c < 3; ++c, ++ti) {
                v8f cz = {};
                v8f d = __builtin_amdgcn_wmma_f32_16x16x4_f32(
                    false, op[r], false, op[c], (short)0, cz, false, false);
#pragma unroll
                for (int q = 0; q < 8; ++q) {
                    float x = d[q];
                    accS[ti][q] += x;
                    accQ[ti][q] += x * x;
                }
            }
    }

    // flush register accumulators: lane owns fixed (i,j) positions
    int ti = 0;
    for (int r = 0; r < 3; ++r)
        for (int c = r; c < 3; ++c, ++ti)
            for (int q = 0; q < 8; ++q) {
                int i = r * 16 + q + half * 8;
                int j = c * 16 + n;
                if (i < j && j < NF) {
                    atomicAdd(stats_sum + i * NF + j, accS[ti][q]);
                    atomicAdd(stats_sq  + i * NF + j, accQ[ti][q]);
                }
            }
}

// ---------------------------------------------------------------------------
// Fold batchnorm + tanh(beta) into per-pair gain G and scalar offset C.
__global__ void fm_finalize_kernel(const float* __restrict__ beta,
                                   const int* __restrict__ feat_i,
                                   const int* __restrict__ feat_j,
                                   const float* __restrict__ stats_sum,
                                   const float* __restrict__ stats_sq,
                                   float* __restrict__ G,
                                   float* __restrict__ Cconst,
                                   int npair, float invB) {
    int p = blockIdx.x * blockDim.x + threadIdx.x;
    if (p >= npair) return;
    int i = feat_i[p], j = feat_j[p];
    int idx = i * NF + j;
    float mean = stats_sum[idx] * invB;
    float var  = stats_sq[idx] * invB - mean * mean;
    float g    = tanhf(beta[p]) * rsqrtf(var + BN_EPS);
    G[idx] = g;
    atomicAdd(Cconst, g * mean);
}

// ---------------------------------------------------------------------------
// Pass 2: out[b] = Σ_{i<j} G[i,j] * <e_i,e_j>  -  C, WMMA tiles + wave reduce.
__global__ void fm_output_kernel(const int* __restrict__ sparse_ids,
                                 const float* __restrict__ dense_vals,
                                 const float* __restrict__ emb_table,
                                 const float* __restrict__ weight,
                                 const float* __restrict__ G,
                                 const float* __restrict__ Cconst,
                                 float* __restrict__ out,
                                 int B, int V) {
    const int lane   = threadIdx.x & 31;
    const int half   = lane >> 4;
    const int n      = lane & 15;
    const int k0     = half * 2;
    const int wave   = (blockIdx.x * blockDim.x + threadIdx.x) >> 5;
    const int nwaves = (gridDim.x * blockDim.x) >> 5;

    float wpre[3];
#pragma unroll
    for (int t = 0; t < 3; ++t) {
        int f = t * 16 + n;
        wpre[t] = (f >= FS && f < NF) ? weight[f - FS] : 0.0f;
    }

    // preload this lane's 48 gain values (zero-padded G makes masking free)
    float gq[6][8];
    {
        int ti = 0;
        for (int r = 0; r < 3; ++r)
            for (int c = r; c < 3; ++c, ++ti)
                for (int q = 0; q < 8; ++q) {
                    int i = r * 16 + q + half * 8;
                    int j = c * 16 + n;
                    gq[ti][q] = (i < NF && j < NF) ? G[i * NF + j] : 0.0f;
                }
    }
    const float Coff = Cconst[0];

    for (int b = wave; b < B; b += nwaves) {
        v2f op[3];
#pragma unroll
        for (int t = 0; t < 3; ++t)
            op[t] = load_operand(sparse_ids, dense_vals, emb_table,
                                 wpre[t], b, t * 16 + n, k0, V);
        float total = 0.0f;
        int ti = 0;
#pragma unroll
        for (int r = 0; r < 3; ++r)
#pragma unroll
            for (int c = r; c < 3; ++c, ++ti) {
                v8f cz = {};
                v8f d = __builtin_amdgcn_wmma_f32_16x16x4_f32(
                    false, op[r], false, op[c], (short)0, cz, false, false);
#pragma unroll
                for (int q = 0; q < 8; ++q) total += gq[ti][q] * d[q];
            }
        // wave32 butterfly reduction
#pragma unroll
        for (int m = 16; m >= 1; m >>= 1) total += __shfl_xor(total, m, 32);
        if (lane == 0) out[b] = total - Coff;
    }
}

// ---------------------------------------------------------------------------
extern "C" void kernel_launch(void* const* d_in, const int* in_sizes, int n_in,
                              void* d_out, int out_size, void* d_ws, size_t ws_size,
                              hipStream_t stream) {
    const int*   sparse_ids = (const int*)  d_in[0];
    const float* dense_vals = (const float*)d_in[1];
    const float* emb_table  = (const float*)d_in[2];
    const float* weight     = (const float*)d_in[3];
    const float* beta       = (const float*)d_in[4];
    const int*   feat_i     = (const int*)  d_in[5];
    const int*   feat_j     = (const int*)  d_in[6];
    float*       out        = (float*)      d_out;

    const int B     = in_sizes[0] / FS;
    const int V     = in_sizes[2] / (FS * EMB_D);
    const int npair = in_sizes[4];

    float* ws        = (float*)d_ws;
    float* stats_sum = ws;
    float* stats_sq  = ws + NSQ;
    float* G         = ws + 2 * NSQ;
    float* Cconst    = ws + 3 * NSQ;

    fm_zero_kernel<<<(WS_FLOATS + 255) / 256, 256, 0, stream>>>(ws);

    // 64 blocks x 256 threads = 512 waves; 32 samples per wave
    fm_stats_kernel<<<64, 256, 0, stream>>>(sparse_ids, dense_vals, emb_table,
                                            weight, stats_sum, stats_sq, B, V);

    fm_finalize_kernel<<<(npair + 255) / 256, 256, 0, stream>>>(
        beta, feat_i, feat_j, stats_sum, stats_sq, G, Cconst, npair,
        1.0f / (float)B);

    // 128 blocks x 256 threads = 1024 waves; 16 samples per wave
    fm_output_kernel<<<128, 256, 0, stream>>>(sparse_ids, dense_vals, emb_table,
                                              weight, G, Cconst, out, B, V);
}